// GeometricExtractor_14912126451982
// MI455X (gfx1250) — compile-verified
//
#include <hip/hip_runtime.h>
#include <hip/hip_bf16.h>
#include <math.h>

#define BB 4
#define NN 8192
#define KNB 9
#define NPTS (BB * NN)          // 32768 points
#define NROWS (BB * NN * KNB)   // 294912 MLP rows
#define NTILES (NN / 16)        // 512 column tiles per row-tile (even)
#define LSTRIDE 20              // LDS gram row stride (floats): bank-clean + 16B aligned

typedef __attribute__((ext_vector_type(2))) float v2f;
typedef __attribute__((ext_vector_type(8))) float v8f;

// ---------------------------------------------------------------------------
// 1) per-point squared norm  sq[p] = |x_p|^2
// ---------------------------------------------------------------------------
__global__ void sq_kernel(const float* __restrict__ x, float* __restrict__ sq) {
  int p = blockIdx.x * blockDim.x + threadIdx.x;
  if (p < NPTS) {
    float a = x[p * 3 + 0], b = x[p * 3 + 1], c = x[p * 3 + 2];
    sq[p] = a * a + b * b + c * c;
  }
}

// sorted-insert into ascending top-10 (ties keep lower index); registers only.
#define TOP10_INSERT(S, NIDX)                                              \
  if ((S) < dlist[9]) {                                                    \
    _Pragma("unroll")                                                      \
    for (int i = 9; i > 0; --i) {                                          \
      if (dlist[i] > (S)) {                                                \
        const bool here = (dlist[i - 1] <= (S));                           \
        dlist[i] = here ? (S) : dlist[i - 1];                              \
        ilist[i] = here ? (NIDX) : ilist[i - 1];                           \
      }                                                                    \
    }                                                                      \
    if (dlist[0] > (S)) { dlist[0] = (S); ilist[0] = (NIDX); }             \
  }

#define CAND(J, GV, SV)                                                    \
  {                                                                        \
    const int   nc = nt * 16 + hf * 8 + (J);                               \
    const float sc = (SV) - 2.0f * (GV);                                   \
    TOP10_INSERT(sc, nc)                                                   \
  }

// ---------------------------------------------------------------------------
// 2) kNN via V_WMMA_F32_16X16X4_F32 gram tiles + per-row top-10 selection.
//    One wave per 16 query rows.  A (16x4 f32): lane=(m=lane&15, half);
//    v0=K(half?2:0), v1=K(half?3:1), K=3 zero-padded. B mirrored on columns.
//    C: vgpr r -> M = r + half*8, N = lane&15.
//    score = sq[n] - 2*gram (sq[m] is per-row constant; same ordering).
//    Tile loop unrolled x2 over two register buffers; the prefetch for
//    tile nt+2 issues right after tile nt's selection (buffer dead -> loads
//    land in the consumer registers, no copies, no premature s_wait) and has
//    a full half-iteration of WMMA/LDS/selection work to hide L2 latency.
// ---------------------------------------------------------------------------
__global__ void __launch_bounds__(32)
knn_kernel(const float* __restrict__ x, const float* __restrict__ sq,
           int* __restrict__ nbr) {
  __shared__ float lgram[16 * LSTRIDE];
  __shared__ float lld[32][10];
  __shared__ int   lli[32][10];

  const int bt   = blockIdx.x;        // [0, BB*NTILES)
  const int b    = bt / NTILES;
  const int tm   = bt % NTILES;
  const int row0 = tm * 16;
  const int lane = threadIdx.x;       // 0..31
  const int ml   = lane & 15;
  const int hf   = lane >> 4;
  const int bN   = b * NN;

  // A operand (resident for the whole scan)
  const float* xa = x + (size_t)(bN + row0 + ml) * 3;
  v2f A;
  A.x = xa[hf * 2];                   // K=0 or K=2
  A.y = hf ? 0.0f : xa[1];            // K=1 or K=3(pad)

  float dlist[10];
  int   ilist[10];
#pragma unroll
  for (int t = 0; t < 10; ++t) { dlist[t] = 3.4e38f; ilist[t] = 0x7fffffff; }

  // tile operand loader (B column strip + 8 contiguous sq values, 2x b128)
  auto load_tile = [&](int nt, v2f& Bv, float4& s0, float4& s1) {
    const int col = nt * 16 + ml;
    const float* xb = x + (size_t)(bN + col) * 3;
    Bv.x = xb[hf * 2];
    Bv.y = hf ? 0.0f : xb[1];
    const float4* sp = (const float4*)(sq + bN + nt * 16 + hf * 8);  // 32B aligned
    s0 = sp[0];
    s1 = sp[1];
  };

  // one tile: WMMA -> LDS stage -> selection; then refill this buffer with
  // tile ntpf (uniform guard keeps EXEC all-ones for the WMMA path).
  auto half_step = [&](int nt, v2f& Bv, float4& s0, float4& s1, int ntpf) {
    v8f C = {};
    C = __builtin_amdgcn_wmma_f32_16x16x4_f32(false, A, false, Bv,
                                              (short)0, C, false, false);
#pragma unroll
    for (int r = 0; r < 8; ++r)
      lgram[(r + hf * 8) * LSTRIDE + ml] = C[r];
    __syncthreads();   // store -> load visibility (single wave: just a DS wait)

    const float4* gp = (const float4*)&lgram[ml * LSTRIDE + hf * 8];
    const float4 g0 = gp[0];
    const float4 g1 = gp[1];

    CAND(0, g0.x, s0.x)  CAND(1, g0.y, s0.y)
    CAND(2, g0.z, s0.z)  CAND(3, g0.w, s0.w)
    CAND(4, g1.x, s1.x)  CAND(5, g1.y, s1.y)
    CAND(6, g1.z, s1.z)  CAND(7, g1.w, s1.w)

    if (ntpf < NTILES) load_tile(ntpf, Bv, s0, s1);  // buffer now dead -> refill
  };

  // prologue: fill both buffers
  v2f B0, B1;
  float4 s00, s01, s10, s11;
  load_tile(0, B0, s00, s01);
  load_tile(1, B1, s10, s11);

  for (int nt = 0; nt < NTILES; nt += 2) {
    half_step(nt,     B0, s00, s01, nt + 2);
    half_step(nt + 1, B1, s10, s11, nt + 3);
  }

  // merge the two half-lists per row (exact top_k tie-break: value, then index)
#pragma unroll
  for (int t = 0; t < 10; ++t) { lld[lane][t] = dlist[t]; lli[lane][t] = ilist[t]; }
  __syncthreads();
  if (lane < 16) {
    int ia = 0, ib = 0;
    int* outp = nbr + (size_t)(bN + row0 + lane) * KNB;
    for (int t = 0; t < 10; ++t) {        // ia+ib == t <= 9, never OOB
      float da = lld[lane][ia],      db = lld[lane + 16][ib];
      int   na = lli[lane][ia],      nb = lli[lane + 16][ib];
      bool  ta = (da < db) || (da == db && na < nb);
      int  sel = ta ? na : nb;
      if (ta) ++ia; else ++ib;
      if (t > 0) outp[t - 1] = sel;       // skip entry 0 (self), positionally
    }
  }
}

// ---------------------------------------------------------------------------
// 3) geometry + layer-1 pre-activations + BN1 partial sums
// ---------------------------------------------------------------------------
__global__ void geom_kernel(const float* __restrict__ x, const int* __restrict__ nbr,
                            const float* __restrict__ W1, const float* __restrict__ b1,
                            float* __restrict__ z1, float* __restrict__ stats) {
  __shared__ float ssum[20];
  if (threadIdx.x < 20) ssum[threadIdx.x] = 0.0f;
  __syncthreads();

  const int p = blockIdx.x * blockDim.x + threadIdx.x;
  float lsum[10], lsq[10];
#pragma unroll
  for (int c = 0; c < 10; ++c) { lsum[c] = 0.0f; lsq[c] = 0.0f; }

  if (p < NPTS) {
    const int b = p / NN;
    const float px = x[p * 3 + 0], py = x[p * 3 + 1], pz = x[p * 3 + 2];

    float rel[9][3], ph[9];
    int ord[9];
    for (int j = 0; j < 9; ++j) {
      const int m = nbr[p * 9 + j];
      const float* xm = x + (size_t)(b * NN + m) * 3;
      rel[j][0] = xm[0] - px;
      rel[j][1] = xm[1] - py;
      rel[j][2] = xm[2] - pz;
      ph[j] = atan2f(rel[j][1], rel[j][0]);
      ord[j] = j;
    }
    // stable insertion sort by phi (matches jnp.argsort stable)
    for (int i = 1; i < 9; ++i) {
      float pv = ph[i]; int ov = ord[i]; int jj = i - 1;
      while (jj >= 0 && ph[jj] > pv) { ph[jj + 1] = ph[jj]; ord[jj + 1] = ord[jj]; --jj; }
      ph[jj + 1] = pv; ord[jj + 1] = ov;
    }
    float v1[9][3];
    for (int i = 0; i < 9; ++i) {
      int o = ord[i];
      v1[i][0] = rel[o][0]; v1[i][1] = rel[o][1]; v1[i][2] = rel[o][2];
    }
    float cen[9][3], nrm[9][3];
    for (int i = 0; i < 9; ++i) {
      int i2 = (i + 1) % 9;  // roll(-1)
      float ax = v1[i][0], ay = v1[i][1], az = v1[i][2];
      float bx = v1[i2][0], by = v1[i2][1], bz = v1[i2][2];
      cen[i][0] = 0.5f * (ax + bx);
      cen[i][1] = 0.5f * (ay + by);
      cen[i][2] = 0.5f * (az + bz);
      float nx = ay * bz - az * by;
      float ny = az * bx - ax * bz;
      float nz = ax * by - ay * bx;
      float inn = 1.0f / (sqrtf(nx * nx + ny * ny + nz * nz) + 1e-6f);
      nrm[i][0] = nx * inn; nrm[i][1] = ny * inn; nrm[i][2] = nz * inn;
    }
    const float sgn = (nrm[0][0] > 0.0f) ? 1.0f : -1.0f;
    const float is3 = 0.57735026918962576f;  // 1/sqrt(3)
    for (int j = 0; j < 9; ++j) {
      float n0 = nrm[j][0] * sgn, n1 = nrm[j][1] * sgn, n2 = nrm[j][2] * sgn;
      float pos = (n0 * cen[j][0] + n1 * cen[j][1] + n2 * cen[j][2]) * is3;
      float feat[7] = {cen[j][0], cen[j][1], cen[j][2], n0, n1, n2, pos};
      float* zr = z1 + (size_t)(p * 9 + j) * 10;
#pragma unroll
      for (int c = 0; c < 10; ++c) {
        float acc = b1[c];
#pragma unroll
        for (int f = 0; f < 7; ++f) acc += feat[f] * W1[f * 10 + c];
        zr[c] = acc;
        lsum[c] += acc;
        lsq[c] += acc * acc;
      }
    }
  }
#pragma unroll
  for (int c = 0; c < 10; ++c) {
    atomicAdd(&ssum[c], lsum[c]);
    atomicAdd(&ssum[10 + c], lsq[c]);
  }
  __syncthreads();
  if (threadIdx.x < 20) atomicAdd(&stats[threadIdx.x], ssum[threadIdx.x]);
}

// ---------------------------------------------------------------------------
// 4) finalize BN stats: bn[0..9]=mu, bn[10..19]=rsqrt(var+eps)
// ---------------------------------------------------------------------------
__global__ void bnfin_kernel(const float* __restrict__ stats, float* __restrict__ bn,
                             float invn) {
  int c = threadIdx.x;
  if (c < 10) {
    float mu  = stats[c] * invn;
    float var = stats[10 + c] * invn - mu * mu;   // biased (jnp.var)
    bn[c] = mu;
    bn[10 + c] = rsqrtf(var + 1e-5f);
  }
}

// ---------------------------------------------------------------------------
// 5) h1 = relu(BN1(z1)); z2 = h1 @ W2 + b2; BN2 partial sums
// ---------------------------------------------------------------------------
__global__ void layer2_kernel(const float* __restrict__ z1, const float* __restrict__ bn1,
                              const float* __restrict__ g1, const float* __restrict__ be1,
                              const float* __restrict__ W2, const float* __restrict__ b2,
                              float* __restrict__ z2, float* __restrict__ stats) {
  __shared__ float ssum[20];
  if (threadIdx.x < 20) ssum[threadIdx.x] = 0.0f;
  __syncthreads();

  const int r = blockIdx.x * blockDim.x + threadIdx.x;
  float lsum[10], lsq[10];
#pragma unroll
  for (int c = 0; c < 10; ++c) { lsum[c] = 0.0f; lsq[c] = 0.0f; }

  if (r < NROWS) {
    float h[10];
#pragma unroll
    for (int f = 0; f < 10; ++f) {
      float v = g1[f] * (z1[(size_t)r * 10 + f] - bn1[f]) * bn1[10 + f] + be1[f];
      h[f] = v > 0.0f ? v : 0.0f;
    }
#pragma unroll
    for (int c = 0; c < 10; ++c) {
      float acc = b2[c];
#pragma unroll
      for (int f = 0; f < 10; ++f) acc += h[f] * W2[f * 10 + c];
      z2[(size_t)r * 10 + c] = acc;
      lsum[c] = acc;
      lsq[c] = acc * acc;
    }
  }
#pragma unroll
  for (int c = 0; c < 10; ++c) {
    atomicAdd(&ssum[c], lsum[c]);
    atomicAdd(&ssum[10 + c], lsq[c]);
  }
  __syncthreads();
  if (threadIdx.x < 20) atomicAdd(&stats[threadIdx.x], ssum[threadIdx.x]);
}

// ---------------------------------------------------------------------------
// 6) h2 = relu(BN2(z2)); out = max over k
// ---------------------------------------------------------------------------
__global__ void final_kernel(const float* __restrict__ z2, const float* __restrict__ bn2,
                             const float* __restrict__ g2, const float* __restrict__ be2,
                             float* __restrict__ out) {
  const int p = blockIdx.x * blockDim.x + threadIdx.x;
  if (p >= NPTS) return;
  float mx[10];
#pragma unroll
  for (int c = 0; c < 10; ++c) mx[c] = -3.4e38f;
  for (int j = 0; j < 9; ++j) {
    const float* zr = z2 + (size_t)(p * 9 + j) * 10;
#pragma unroll
    for (int c = 0; c < 10; ++c) {
      float v = g2[c] * (zr[c] - bn2[c]) * bn2[10 + c] + be2[c];
      v = v > 0.0f ? v : 0.0f;
      mx[c] = fmaxf(mx[c], v);
    }
  }
#pragma unroll
  for (int c = 0; c < 10; ++c) out[(size_t)p * 10 + c] = mx[c];
}

// ---------------------------------------------------------------------------
extern "C" void kernel_launch(void* const* d_in, const int* in_sizes, int n_in,
                              void* d_out, int out_size, void* d_ws, size_t ws_size,
                              hipStream_t stream) {
  (void)in_sizes; (void)n_in; (void)out_size; (void)ws_size;
  const float* x   = (const float*)d_in[0];
  const float* W1  = (const float*)d_in[1];
  const float* b1  = (const float*)d_in[2];
  const float* g1  = (const float*)d_in[3];
  const float* be1 = (const float*)d_in[4];
  const float* W2  = (const float*)d_in[5];
  const float* b2  = (const float*)d_in[6];
  const float* g2  = (const float*)d_in[7];
  const float* be2 = (const float*)d_in[8];

  // workspace layout (floats): sq | nbr(int) | z1 | z2 | st1 | st2 | bn1 | bn2
  float* ws  = (float*)d_ws;
  float* sq  = ws;
  int*   nbr = (int*)(ws + NPTS);
  float* z1  = ws + NPTS + NPTS * KNB;
  float* z2  = z1 + (size_t)NROWS * 10;
  float* st1 = z2 + (size_t)NROWS * 10;
  float* st2 = st1 + 20;
  float* bn1 = st2 + 20;
  float* bn2 = bn1 + 20;

  hipMemsetAsync(st1, 0, 40 * sizeof(float), stream);  // st1 + st2

  sq_kernel<<<(NPTS + 255) / 256, 256, 0, stream>>>(x, sq);
  knn_kernel<<<BB * NTILES, 32, 0, stream>>>(x, sq, nbr);
  geom_kernel<<<(NPTS + 255) / 256, 256, 0, stream>>>(x, nbr, W1, b1, z1, st1);
  bnfin_kernel<<<1, 32, 0, stream>>>(st1, bn1, 1.0f / (float)NROWS);
  layer2_kernel<<<(NROWS + 255) / 256, 256, 0, stream>>>(z1, bn1, g1, be1, W2, b2, z2, st2);
  bnfin_kernel<<<1, 32, 0, stream>>>(st2, bn2, 1.0f / (float)NROWS);
  final_kernel<<<(NPTS + 255) / 256, 256, 0, stream>>>(z2, bn2, g2, be2, (float*)d_out);
}